// Random_GCN_27290222198916
// MI455X (gfx1250) — compile-verified
//
#include <hip/hip_runtime.h>
#include <hip/hip_bf16.h>
#include <math.h>

// GCN forward on MI455X (gfx1250), wave32 + V_WMMA_F32_16X16X4_F32.
//
// Inputs (setup_inputs order):
//  0: x    [8192,512] f32      4: W1 [512,128] f32
//  1: adj  [8192,8192] f32     5: b1 [128] f32
//  2: m1   [8192,8192] f32     6: W2 [128,16] f32
//  3: m2   [8192,8192] f32     7: b2 [16] f32
// Output: log_softmax(a@(relu(a@(x@W1)+b1)@W2)+b2)  [8192,16] f32
//
// Roofline: dominated by streaming the 8192^2 adjacency/mask arrays.
// If ws_size allows, the degree pass materializes a = adj*m1*m2 (268 MB)
// so layers 1/2 stream one array instead of three:  ~1.6 GB total HBM
// traffic (vs 2.4 GB recompute fallback) at 23.3 TB/s.

#define NN    8192
#define NF    512
#define NH    128
#define NC    16
#define KC    64     // K-chunk staged through LDS

typedef __attribute__((ext_vector_type(2))) float v2f;
typedef __attribute__((ext_vector_type(4))) float f32x4;
typedef __attribute__((ext_vector_type(8))) float v8f;

__device__ __forceinline__ f32x4 nt_load4(const float* p) {
    return __builtin_nontemporal_load((const f32x4*)p);
}
__device__ __forceinline__ void nt_store4(float* p, f32x4 v) {
    __builtin_nontemporal_store(v, (f32x4*)p);
}

// ---------------------------------------------------------------------------
// K1: deg[row] = sum_j adj*m1*m2 ; dinv[row] = deg>0 ? deg^-1/2 : 0.
// CACHED: also materialize aw[row][:] = adj*m1*m2 (NT store, read later).
// One block (256 threads) per row; 128-bit NT streaming loads.
// ---------------------------------------------------------------------------
template <bool CACHED>
__global__ void __launch_bounds__(256)
gcn_degree(const float* __restrict__ adj, const float* __restrict__ m1,
           const float* __restrict__ m2, float* __restrict__ dinv,
           float* __restrict__ aw) {
    const int row = blockIdx.x;
    const size_t base = (size_t)row * NN;
    float s = 0.f;
    for (int j = threadIdx.x * 4; j < NN; j += 256 * 4) {
        f32x4 a = nt_load4(adj + base + j);
        f32x4 b = nt_load4(m1 + base + j);
        f32x4 c = nt_load4(m2 + base + j);
        f32x4 p;
        p.x = a.x * b.x * c.x; p.y = a.y * b.y * c.y;
        p.z = a.z * b.z * c.z; p.w = a.w * b.w * c.w;
        if (CACHED) nt_store4(aw + base + j, p);
        s += p.x + p.y + p.z + p.w;
    }
    for (int off = 16; off > 0; off >>= 1) s += __shfl_down(s, off, 32);
    __shared__ float red[8];
    const int lane = threadIdx.x & 31, wid = threadIdx.x >> 5;
    if (lane == 0) red[wid] = s;
    __syncthreads();
    if (threadIdx.x == 0) {
        float d = 0.f;
        for (int i = 0; i < 8; ++i) d += red[i];
        dinv[row] = (d > 0.f) ? (1.0f / sqrtf(d)) : 0.f;
    }
}

// ---------------------------------------------------------------------------
// K2: XW1 = x @ W1   [8192,512]@[512,128]
// Block: 16 M-rows x 128 N (8 waves, one 16x16 WMMA tile each), K in LDS.
// ---------------------------------------------------------------------------
__global__ void __launch_bounds__(256)
gcn_xw1(const float* __restrict__ x, const float* __restrict__ W1,
        float* __restrict__ XW1) {
    __shared__ float As[16][68];    // padded -> conflict-free frag reads
    __shared__ float Bs[KC][132];
    const int rowBase = blockIdx.x * 16;
    const int t = threadIdx.x;
    const int wave = t >> 5, lane = t & 31;
    const int m = lane & 15, half = lane >> 4;
    const int lr = t >> 4, lc = (t & 15) * 4;

    v8f acc = {};
    for (int k = 0; k < NF; k += KC) {
        {
            f32x4 v = nt_load4(x + (size_t)(rowBase + lr) * NF + k + lc);
            As[lr][lc] = v.x; As[lr][lc + 1] = v.y; As[lr][lc + 2] = v.z; As[lr][lc + 3] = v.w;
        }
        for (int j = 0; j < 8; ++j) {
            int idx = t + j * 256;
            int r = idx >> 5, c = (idx & 31) * 4;
            f32x4 v = *(const f32x4*)(W1 + (size_t)(k + r) * NH + c);
            Bs[r][c] = v.x; Bs[r][c + 1] = v.y; Bs[r][c + 2] = v.z; Bs[r][c + 3] = v.w;
        }
        __syncthreads();
        for (int kk = 0; kk < KC; kk += 4) {
            const int k0 = kk + 2 * half;
            v2f a; a.x = As[m][k0];              a.y = As[m][k0 + 1];
            v2f b; b.x = Bs[k0][wave * 16 + m];  b.y = Bs[k0 + 1][wave * 16 + m];
            acc = __builtin_amdgcn_wmma_f32_16x16x4_f32(false, a, false, b,
                                                        (short)0, acc, false, false);
        }
        __syncthreads();
    }
    for (int v = 0; v < 8; ++v)
        XW1[(size_t)(rowBase + v + 8 * half) * NH + wave * 16 + m] = acc[v];
}

// ---------------------------------------------------------------------------
// K3: layer 1 fused.  Per 32-row M-tile (2 accumulators/wave share each B
// fragment -> halves XW1 L2 traffic, 2 back-to-back WMMAs per frag pair):
//   Anorm tile built on the fly, acc += Anorm @ XW1,
//   H = relu(acc + b1) kept in LDS, HW2 = H @ W2 stored (32x16).
// ---------------------------------------------------------------------------
template <bool CACHED>
__global__ void __launch_bounds__(256)
gcn_layer1(const float* __restrict__ adj, const float* __restrict__ m1,
           const float* __restrict__ m2, const float* __restrict__ aw,
           const float* __restrict__ dinv, const float* __restrict__ XW1,
           const float* __restrict__ b1, const float* __restrict__ W2,
           float* __restrict__ HW2) {
    __shared__ float As[32][68];
    __shared__ float Bs[KC][132];
    __shared__ float Hs[32][136];
    const int rowBase = blockIdx.x * 32;
    const int t = threadIdx.x;
    const int wave = t >> 5, lane = t & 31;
    const int m = lane & 15, half = lane >> 4;
    const int lr = t >> 4, lc = (t & 15) * 4;
    const float drow0 = dinv[rowBase + lr];
    const float drow1 = dinv[rowBase + lr + 16];

    v8f acc0 = {}, acc1 = {};
    for (int k = 0; k < NN; k += KC) {
        { // stage normalized-adjacency tile, rows lr and lr+16
            const f32x4 dc = *(const f32x4*)(dinv + k + lc);
            const size_t g0 = (size_t)(rowBase + lr) * NN + k + lc;
            const size_t g1 = g0 + (size_t)16 * NN;
            f32x4 p0, p1;
            if (CACHED) {
                p0 = nt_load4(aw + g0);
                p1 = nt_load4(aw + g1);
            } else {
                f32x4 a = nt_load4(adj + g0), b = nt_load4(m1 + g0), c = nt_load4(m2 + g0);
                p0.x = a.x * b.x * c.x; p0.y = a.y * b.y * c.y;
                p0.z = a.z * b.z * c.z; p0.w = a.w * b.w * c.w;
                a = nt_load4(adj + g1); b = nt_load4(m1 + g1); c = nt_load4(m2 + g1);
                p1.x = a.x * b.x * c.x; p1.y = a.y * b.y * c.y;
                p1.z = a.z * b.z * c.z; p1.w = a.w * b.w * c.w;
            }
            As[lr][lc]          = p0.x * drow0 * dc.x;
            As[lr][lc + 1]      = p0.y * drow0 * dc.y;
            As[lr][lc + 2]      = p0.z * drow0 * dc.z;
            As[lr][lc + 3]      = p0.w * drow0 * dc.w;
            As[lr + 16][lc]     = p1.x * drow1 * dc.x;
            As[lr + 16][lc + 1] = p1.y * drow1 * dc.y;
            As[lr + 16][lc + 2] = p1.z * drow1 * dc.z;
            As[lr + 16][lc + 3] = p1.w * drow1 * dc.w;
        }
        for (int j = 0; j < 8; ++j) { // stage B: XW1[k..k+KC][0..128] (L2-resident)
            int idx = t + j * 256;
            int r = idx >> 5, c = (idx & 31) * 4;
            f32x4 v = *(const f32x4*)(XW1 + (size_t)(k + r) * NH + c);
            Bs[r][c] = v.x; Bs[r][c + 1] = v.y; Bs[r][c + 2] = v.z; Bs[r][c + 3] = v.w;
        }
        __syncthreads();
        for (int kk = 0; kk < KC; kk += 4) {
            const int k0 = kk + 2 * half;
            v2f b;  b.x  = Bs[k0][wave * 16 + m]; b.y  = Bs[k0 + 1][wave * 16 + m];
            v2f a0; a0.x = As[m][k0];             a0.y = As[m][k0 + 1];
            v2f a1; a1.x = As[m + 16][k0];        a1.y = As[m + 16][k0 + 1];
            acc0 = __builtin_amdgcn_wmma_f32_16x16x4_f32(false, a0, false, b,
                                                         (short)0, acc0, false, false);
            acc1 = __builtin_amdgcn_wmma_f32_16x16x4_f32(false, a1, false, b,
                                                         (short)0, acc1, false, false);
        }
        __syncthreads();
    }
    // bias + relu -> Hs (32 x 128 across 8 waves)
    for (int v = 0; v < 8; ++v) {
        const int r = v + 8 * half, c = wave * 16 + m;
        float h0 = acc0[v] + b1[c];
        float h1 = acc1[v] + b1[c];
        Hs[r][c]      = h0 > 0.f ? h0 : 0.f;
        Hs[r + 16][c] = h1 > 0.f ? h1 : 0.f;
    }
    __syncthreads();
    // fused HW2 = H(32x128) @ W2(128x16): two output elements per thread
    const int orow = t >> 4, cls = t & 15;
    float s0 = 0.f, s1 = 0.f;
    for (int kk = 0; kk < NH; ++kk) {
        const float w = W2[kk * NC + cls];
        s0 += Hs[orow][kk] * w;
        s1 += Hs[orow + 16][kk] * w;
    }
    HW2[(size_t)(rowBase + orow) * NC + cls]      = s0;
    HW2[(size_t)(rowBase + orow + 16) * NC + cls] = s1;
}

// ---------------------------------------------------------------------------
// K4: out = log_softmax(Anorm @ HW2 + b2).  16-row M-tile per block; 8 waves
// split each KC chunk (8 K each -> 2 WMMA per wave per chunk); partials
// reduced through LDS; exact log-softmax over the 16 classes.
// ---------------------------------------------------------------------------
template <bool CACHED>
__global__ void __launch_bounds__(256)
gcn_layer2(const float* __restrict__ adj, const float* __restrict__ m1,
           const float* __restrict__ m2, const float* __restrict__ aw,
           const float* __restrict__ dinv, const float* __restrict__ HW2,
           const float* __restrict__ b2, float* __restrict__ out) {
    __shared__ float As[16][68];
    __shared__ float Bs[KC][20];
    __shared__ float Red[8][16][17];
    __shared__ float tile[16][17];
    __shared__ float rmax[16], rlse[16];
    const int rowBase = blockIdx.x * 16;
    const int t = threadIdx.x;
    const int wave = t >> 5, lane = t & 31;
    const int m = lane & 15, half = lane >> 4;
    const int lr = t >> 4, lc = (t & 15) * 4;
    const float drow = dinv[rowBase + lr];

    v8f acc = {};
    for (int k = 0; k < NN; k += KC) {
        { // stage normalized-adjacency tile
            const size_t g = (size_t)(rowBase + lr) * NN + k + lc;
            const f32x4 dc = *(const f32x4*)(dinv + k + lc);
            f32x4 p;
            if (CACHED) {
                p = nt_load4(aw + g);
            } else {
                f32x4 a = nt_load4(adj + g), b = nt_load4(m1 + g), c = nt_load4(m2 + g);
                p.x = a.x * b.x * c.x; p.y = a.y * b.y * c.y;
                p.z = a.z * b.z * c.z; p.w = a.w * b.w * c.w;
            }
            As[lr][lc]     = p.x * drow * dc.x;
            As[lr][lc + 1] = p.y * drow * dc.y;
            As[lr][lc + 2] = p.z * drow * dc.z;
            As[lr][lc + 3] = p.w * drow * dc.w;
        }
        { // stage B: HW2[k..k+KC][0..16] = 256 float4
            const int r = t >> 2, c = (t & 3) * 4;
            f32x4 v = *(const f32x4*)(HW2 + (size_t)(k + r) * NC + c);
            Bs[r][c] = v.x; Bs[r][c + 1] = v.y; Bs[r][c + 2] = v.z; Bs[r][c + 3] = v.w;
        }
        __syncthreads();
        for (int s = 0; s < 2; ++s) {   // wave covers K sub-range [wave*8, wave*8+8)
            const int k0 = wave * 8 + s * 4 + 2 * half;
            v2f a; a.x = As[m][k0];  a.y = As[m][k0 + 1];
            v2f b; b.x = Bs[k0][m];  b.y = Bs[k0 + 1][m];
            acc = __builtin_amdgcn_wmma_f32_16x16x4_f32(false, a, false, b,
                                                        (short)0, acc, false, false);
        }
        __syncthreads();
    }
    for (int v = 0; v < 8; ++v) Red[wave][v + 8 * half][m] = acc[v];
    __syncthreads();
    const int orow = t >> 4, cls = t & 15;
    float val = b2[cls];
    for (int w = 0; w < 8; ++w) val += Red[w][orow][cls];
    tile[orow][cls] = val;
    __syncthreads();
    if (t < 16) {
        float mx = tile[t][0];
        for (int j = 1; j < NC; ++j) mx = fmaxf(mx, tile[t][j]);
        float se = 0.f;
        for (int j = 0; j < NC; ++j) se += expf(tile[t][j] - mx);
        rmax[t] = mx;
        rlse[t] = logf(se);
    }
    __syncthreads();
    out[(size_t)(rowBase + orow) * NC + cls] = val - rmax[orow] - rlse[orow];
}

// ---------------------------------------------------------------------------
extern "C" void kernel_launch(void* const* d_in, const int* in_sizes, int n_in,
                              void* d_out, int out_size, void* d_ws, size_t ws_size,
                              hipStream_t stream) {
    const float* x   = (const float*)d_in[0];
    const float* adj = (const float*)d_in[1];
    const float* m1  = (const float*)d_in[2];
    const float* m2  = (const float*)d_in[3];
    const float* W1  = (const float*)d_in[4];
    const float* b1  = (const float*)d_in[5];
    const float* W2  = (const float*)d_in[6];
    const float* b2  = (const float*)d_in[7];
    float* outp = (float*)d_out;

    float* dinv = (float*)d_ws;                           // 8192 floats
    float* XW1  = dinv + NN;                              // 8192*128 floats
    float* HW2  = XW1 + (size_t)NN * NH;                  // 8192*16 floats
    size_t awOff = ((size_t)NN + (size_t)NN * NH + (size_t)NN * NC + 63) & ~(size_t)63;
    float* aw = (float*)d_ws + awOff;                     // 8192*8192 floats (optional)
    const size_t needBytes = (awOff + (size_t)NN * NN) * sizeof(float);
    const bool cached = ws_size >= needBytes;             // deterministic per run

    if (cached) {
        gcn_degree<true> <<<dim3(NN),      dim3(256), 0, stream>>>(adj, m1, m2, dinv, aw);
        gcn_xw1          <<<dim3(NN / 16), dim3(256), 0, stream>>>(x, W1, XW1);
        gcn_layer1<true> <<<dim3(NN / 32), dim3(256), 0, stream>>>(adj, m1, m2, aw, dinv, XW1, b1, W2, HW2);
        gcn_layer2<true> <<<dim3(NN / 16), dim3(256), 0, stream>>>(adj, m1, m2, aw, dinv, HW2, b2, outp);
    } else {
        gcn_degree<false> <<<dim3(NN),      dim3(256), 0, stream>>>(adj, m1, m2, dinv, aw);
        gcn_xw1           <<<dim3(NN / 16), dim3(256), 0, stream>>>(x, W1, XW1);
        gcn_layer1<false> <<<dim3(NN / 32), dim3(256), 0, stream>>>(adj, m1, m2, aw, dinv, XW1, b1, W2, HW2);
        gcn_layer2<false> <<<dim3(NN / 16), dim3(256), 0, stream>>>(adj, m1, m2, aw, dinv, HW2, b2, outp);
    }
}